// myConv_51281909514772
// MI455X (gfx1250) — compile-verified
//
#include <hip/hip_runtime.h>
#include <stdint.h>

// ---------------------------------------------------------------------------
// Implicit-GEMM 3x3 conv (stride 1, pad 1) for MI455X / gfx1250.
//   x: (32, 256, 56, 56) f32   W: (256, 256, 3, 3) f32   out: (32, 256, 56, 56)
// GEMM: M = 100352, N = 256, K = 2304 iterated as (tap, ic): 9 taps x 8
// stages of 32 input channels. W pre-transposed once into d_ws as
// Wt[oc][tap][ic] so each B tile is a clean 2D tile for the Tensor Data
// Mover (one tensor_load_to_lds per stage, TENSORcnt-tracked, LDS padding
// done by the TDM pad feature). A tile im2col'd with coalesced loads.
// Double-buffered LDS, 2x2 register blocking: block tile 128(M) x 64(N),
// each wave 32x32 via 4 accumulators. fp32 WMMA (V_WMMA_F32_16X16X4_F32).
// ---------------------------------------------------------------------------

typedef float v2f __attribute__((ext_vector_type(2)));
typedef float v8f __attribute__((ext_vector_type(8)));
typedef unsigned int v4u __attribute__((ext_vector_type(4)));
typedef int v4i __attribute__((ext_vector_type(4)));
typedef int v8i __attribute__((ext_vector_type(8)));

#define IMG_H   56
#define IMG_W   56
#define HW      3136
#define CIN     256
#define COUT    256
#define NBATCH  32
#define NTAP    9
#define KTOT    2304
#define TK      32
#define NSTAGE  72
#define TM      128
#define TN      64
#define A_PAD   34            // conflict-free col writes, 8B-aligned float2 reads
#define B_PAD   36            // 36-float rows produced by TDM padding
#define WT_ELEMS (COUT * NTAP * CIN)          // 589824
#define WT_BYTES (WT_ELEMS * sizeof(float))   // 2359296
#define B_BUF_BYTES (TN * B_PAD * 4)          // 9216

// ---------------- W transpose: OIHW -> [oc][tap][ic] (write-coalesced) ------
__global__ __launch_bounds__(256)
void transpose_w_kernel(const float* __restrict__ w, float* __restrict__ wt)
{
    int idx = blockIdx.x * 256 + threadIdx.x;
    if (idx < WT_ELEMS) {
        int oc  = idx / (NTAP * CIN);
        int r   = idx - oc * (NTAP * CIN);
        int tap = r >> 8;
        int ic  = r & 255;
        wt[idx] = w[(oc * CIN + ic) * NTAP + tap];
    }
}

// ---- TDM: load one 64x32-f32 B tile (row stride 2304 f32) into LDS,
//      padding 4 DWORDs after every 32 DWORDs -> 36-float LDS rows. ----
__device__ __forceinline__ void tdm_load_b_tile(uint32_t lds_addr, const float* gptr)
{
    uint64_t ga = (uint64_t)(uintptr_t)gptr;
    v4u g0;
    g0.x = 1u;                                           // count=1, user desc
    g0.y = lds_addr;                                     // bits 63:32
    g0.z = (uint32_t)ga;                                 // global_addr[31:0]
    g0.w = (uint32_t)((ga >> 32) & 0x01FFFFFFu)          // global_addr[56:32]
         | (2u << 30);                                   // type = 2 (image)
    v8i g1;
    g1[0] = (int)((2u << 16)      // data_size = 4B
                | (1u << 20)      // pad_enable
                | (4u << 22)      // pad_interval: 32 DWORDs
                | (3u << 25));    // pad_amount: 4 DWORDs
    g1[1] = (int)((KTOT & 0xFFFFu) << 16);               // tensor_dim0 lo16 @63:48
    g1[2] = (int)(((KTOT >> 16) & 0xFFFFu)               // tensor_dim0 hi16
                | ((4096u & 0xFFFFu) << 16));            // tensor_dim1 lo16 (no OOB)
    g1[3] = (int)(0u | (32u << 16));                     // tile_dim0 = 32 @127:112
    g1[4] = (int)(64u);                                  // tile_dim1 = 64, tile_dim2 = 0
    g1[5] = (int)(KTOT);                                 // tensor_dim0_stride[31:0]
    g1[6] = 0;                                           // stride hi, dim1_stride lo
    g1[7] = 0;
    v4i z4 = {0, 0, 0, 0};
    v8i z8 = {0, 0, 0, 0, 0, 0, 0, 0};
    // 6-arg toolchain variant: (g0, g1, g2, g3, extra, cpol)
    __builtin_amdgcn_tensor_load_to_lds(g0, g1, z4, z4, z8, 0);
}

// ---------------- main kernel: tap-ordered, double-buffered, TDM B ----------
__global__ __launch_bounds__(256)
void conv3x3_wmma_tdm(const float* __restrict__ xin,
                      const float* __restrict__ wt,
                      float* __restrict__ out)
{
    __shared__ float As[2][TM][A_PAD];   // [buf][m][ic]
    __shared__ float Bs[2][TN][B_PAD];   // [buf][oc][ic]  (rows written by TDM)

    const int tid    = threadIdx.x;
    const int mBase  = blockIdx.x * TM;
    const int ocBase = blockIdx.y * TN;

    // ---- A loader: one output pixel per thread (m = tid&127), 16 ic each ----
    const int mA  = tid & 127;
    const int icq = (tid >> 7) << 4;                 // 0 or 16
    const int p   = mBase + mA;
    const int pn  = p / HW;
    const int pr  = p - pn * HW;
    const int pyv = pr / IMG_W;
    const int pxv = pr - pyv * IMG_W;
    const float* xbase = xin + (size_t)pn * CIN * HW;

    // ---- B tile base (row 0 of this block's oc range) ----
    const float* wtile = wt + (size_t)(ocBase * NTAP) * CIN;
    const uint32_t ldsB = (uint32_t)(uintptr_t)&Bs[0][0][0];

    // ---- compute mapping: 8 waves -> 4(M) x 2(N); each wave 32x32 ----
    const int lane = tid & 31;
    const int wave = tid >> 5;
    const int m0   = (wave >> 1) << 5;               // 0,32,64,96
    const int n0   = (wave & 1) << 5;                // 0,32
    const int mi   = lane & 15;
    const int kk   = (lane >> 4) << 1;               // K+0/+1 vs K+2/+3
    const bool lead = (__builtin_amdgcn_readfirstlane(tid) == 0);  // wave 0 only

    v8f acc00 = {}, acc01 = {}, acc10 = {}, acc11 = {};

    // ---- prologue: stage 0 (tap 0 -> kh=0,kw=0; icBase=0) ----
    if (lead)
        tdm_load_b_tile(ldsB, wtile);
    {
        const int iy = pyv - 1, ix = pxv - 1;
        const bool ok = ((unsigned)iy < (unsigned)IMG_H) &&
                        ((unsigned)ix < (unsigned)IMG_W);
        const float* src = xbase + icq * HW + iy * IMG_W + ix;
#pragma unroll
        for (int j = 0; j < 16; ++j)
            As[0][mA][icq + j] = ok ? src[j * HW] : 0.0f;
    }

    for (int s = 0; s < NSTAGE; ++s) {
        const int cur = s & 1;
        const int nxt = cur ^ 1;

        float ar[16];
        if (s + 1 < NSTAGE) {
            const int s1     = s + 1;
            const int tap    = s1 >> 3;
            const int icBase = (s1 & 7) << 5;
            const int off    = tap * CIN + icBase;

            if (lead)
                tdm_load_b_tile(ldsB + (uint32_t)(nxt * B_BUF_BYTES), wtile + off);

            const int kh = tap / 3;
            const int kw = tap - kh * 3;
            const int iy = pyv + kh - 1;
            const int ix = pxv + kw - 1;
            const bool ok = ((unsigned)iy < (unsigned)IMG_H) &&
                            ((unsigned)ix < (unsigned)IMG_W);
            const float* src = xbase + (icBase + icq) * HW + iy * IMG_W + ix;
#pragma unroll
            for (int j = 0; j < 16; ++j)
                ar[j] = ok ? src[j * HW] : 0.0f;

            if (lead)
                __builtin_amdgcn_s_wait_tensorcnt(1);  // stage-s tile landed
        } else {
            if (lead)
                __builtin_amdgcn_s_wait_tensorcnt(0);
        }
        __syncthreads();   // B(s) visible to all; A(s) stores visible

        // ---- 8 K-steps of 4: fp32 WMMA, 2x2 register blocking ----
#pragma unroll
        for (int kc = 0; kc < TK; kc += 4) {
            v2f a0 = *(const v2f*)&As[cur][m0 + mi][kc + kk];
            v2f a1 = *(const v2f*)&As[cur][m0 + 16 + mi][kc + kk];
            v2f b0 = *(const v2f*)&Bs[cur][n0 + mi][kc + kk];
            v2f b1 = *(const v2f*)&Bs[cur][n0 + 16 + mi][kc + kk];
            acc00 = __builtin_amdgcn_wmma_f32_16x16x4_f32(
                        false, a0, false, b0, (short)0, acc00, false, false);
            acc01 = __builtin_amdgcn_wmma_f32_16x16x4_f32(
                        false, a0, false, b1, (short)0, acc01, false, false);
            acc10 = __builtin_amdgcn_wmma_f32_16x16x4_f32(
                        false, a1, false, b0, (short)0, acc10, false, false);
            acc11 = __builtin_amdgcn_wmma_f32_16x16x4_f32(
                        false, a1, false, b1, (short)0, acc11, false, false);
        }

        if (s + 1 < NSTAGE) {
#pragma unroll
            for (int j = 0; j < 16; ++j)
                As[nxt][mA][icq + j] = ar[j];
        }
        __syncthreads();   // all waves done reading cur before reuse
    }

    // ---- store D (VGPR r -> M = r or r+8; lane%16 -> N) to NCHW ----
    const int hi  = (lane >> 4) << 3;                // 0 or 8
    const int oc0 = ocBase + n0 + mi;
#pragma unroll
    for (int r = 0; r < 8; ++r) {
        int p0 = mBase + m0 + hi + r;
        int n  = p0 / HW;
        int yx = p0 - n * HW;
        out[(n * COUT + oc0) * HW + yx]      = acc00[r];
        out[(n * COUT + oc0 + 16) * HW + yx] = acc01[r];
        int p1 = p0 + 16;
        n  = p1 / HW;
        yx = p1 - n * HW;
        out[(n * COUT + oc0) * HW + yx]      = acc10[r];
        out[(n * COUT + oc0 + 16) * HW + yx] = acc11[r];
    }
}

// ---------------- fallback (no workspace): per-lane async version -----------
__global__ __launch_bounds__(256)
void conv3x3_wmma_f32_fallback(const float* __restrict__ xin,
                               const float* __restrict__ wgt,
                               float* __restrict__ out)
{
    __shared__ float As[64][B_PAD];
    __shared__ float Bs[64][B_PAD];

    const int tid    = threadIdx.x;
    const int mBase  = blockIdx.x * 64;
    const int ocBase = blockIdx.y * 64;

    const int kcol  = tid & 31;
    const int mrow0 = tid >> 5;
    int pn[8], py[8], px[8];
#pragma unroll
    for (int j = 0; j < 8; ++j) {
        int pp = mBase + mrow0 + 8 * j;
        int n = pp / HW;
        int r = pp - n * HW;
        int y = r / IMG_W;
        pn[j] = n; py[j] = y; px[j] = r - y * IMG_W;
    }

    const int f0 = tid, f1 = tid + 256;
    const int bn0 = f0 >> 3, bk0 = (f0 & 7) << 2;
    const int bn1 = f1 >> 3, bk1 = (f1 & 7) << 2;
    const uint32_t ldsB0 = (uint32_t)(uintptr_t)&Bs[bn0][bk0];
    const uint32_t ldsB1 = (uint32_t)(uintptr_t)&Bs[bn1][bk1];
    const float* wrow0 = wgt + (ocBase + bn0) * KTOT + bk0;
    const float* wrow1 = wgt + (ocBase + bn1) * KTOT + bk1;

    const int lane = tid & 31;
    const int wave = tid >> 5;
    const int m0   = (wave >> 1) << 4;
    const int n0   = (wave & 1) << 5;
    const int mi   = lane & 15;
    const int kk   = (lane >> 4) << 1;

    v8f acc0 = {};
    v8f acc1 = {};

    for (int s = 0; s < KTOT / TK; ++s) {
        const int kBase = s * TK;
        __syncthreads();
        {
            uint64_t ga0 = (uint64_t)(uintptr_t)(wrow0 + kBase);
            uint64_t ga1 = (uint64_t)(uintptr_t)(wrow1 + kBase);
            asm volatile("global_load_async_to_lds_b128 %0, %1, off"
                         :: "v"(ldsB0), "v"(ga0) : "memory");
            asm volatile("global_load_async_to_lds_b128 %0, %1, off"
                         :: "v"(ldsB1), "v"(ga1) : "memory");
        }
        {
            int kg = kBase + kcol;
            int ic = kg / 9;
            int r9 = kg - ic * 9;
            int kh = r9 / 3;
            int kw = r9 - kh * 3;
#pragma unroll
            for (int j = 0; j < 8; ++j) {
                int iy = py[j] + kh - 1;
                int ix = px[j] + kw - 1;
                float v = 0.0f;
                if ((unsigned)iy < (unsigned)IMG_H && (unsigned)ix < (unsigned)IMG_W)
                    v = xin[((pn[j] * CIN + ic) * IMG_H + iy) * IMG_W + ix];
                As[mrow0 + 8 * j][kcol] = v;
            }
        }
        asm volatile("s_wait_asynccnt 0" ::: "memory");
        __syncthreads();
#pragma unroll
        for (int kc = 0; kc < TK; kc += 4) {
            v2f a  = *(const v2f*)&As[m0 + mi][kc + kk];
            v2f b0 = *(const v2f*)&Bs[n0 + mi][kc + kk];
            v2f b1 = *(const v2f*)&Bs[n0 + 16 + mi][kc + kk];
            acc0 = __builtin_amdgcn_wmma_f32_16x16x4_f32(
                       false, a, false, b0, (short)0, acc0, false, false);
            acc1 = __builtin_amdgcn_wmma_f32_16x16x4_f32(
                       false, a, false, b1, (short)0, acc1, false, false);
        }
    }

    const int hi  = (lane >> 4) << 3;
    const int oc0 = ocBase + n0 + mi;
#pragma unroll
    for (int r = 0; r < 8; ++r) {
        int pp = mBase + m0 + hi + r;
        int n  = pp / HW;
        int yx = pp - n * HW;
        out[(n * COUT + oc0) * HW + yx]      = acc0[r];
        out[(n * COUT + oc0 + 16) * HW + yx] = acc1[r];
    }
}

extern "C" void kernel_launch(void* const* d_in, const int* in_sizes, int n_in,
                              void* d_out, int out_size, void* d_ws, size_t ws_size,
                              hipStream_t stream)
{
    (void)in_sizes; (void)n_in; (void)out_size;
    const float* x = (const float*)d_in[0];
    const float* w = (const float*)d_in[1];
    float* out     = (float*)d_out;

    if (d_ws != nullptr && ws_size >= (size_t)WT_BYTES) {
        float* wt = (float*)d_ws;
        transpose_w_kernel<<<dim3((WT_ELEMS + 255) / 256), dim3(256), 0, stream>>>(w, wt);
        dim3 grid((NBATCH * HW) / TM, COUT / TN);      // 784 x 4
        conv3x3_wmma_tdm<<<grid, dim3(256), 0, stream>>>(x, wt, out);
    } else {
        dim3 grid((NBATCH * HW) / 64, COUT / 64);      // 1568 x 4
        conv3x3_wmma_f32_fallback<<<grid, dim3(256), 0, stream>>>(x, w, out);
    }
}